// sknn_model_37675453120699
// MI455X (gfx1250) — compile-verified
//
#include <hip/hip_runtime.h>
#include <hip/hip_bf16.h>

// RBF-KNN localization, fused.  N=4096 queries, M=50000 refs, D=5.
// WMMA directly produces scale*d2 (sigma scale + both norm terms folded into
// the GEMM via spare K slots, C = inline 0):
//   K0..4 : A = -2*scale*q[k],  B = r[k]
//   K5    : A = scale*|q|^2,    B = 1
//   K6    : A = 1,              B = scale*|r|^2
//   K7    : 0
// w = exp2(scale*d2) directly (no clamp: d2 can only dip ~1e-5 below 0 from
// f32 rounding -> w off by ~2e-6 relative, cancelled by the w/sum(w) norm),
// then per-lane FMA accumulation + shfl_xor butterfly.

#define NQ 4096
#define MR 50000
#define DD 5
#define QT (NQ / 16)   // 256 query tiles
#define RT (MR / 16)   // 3125 ref tiles

typedef __attribute__((ext_vector_type(2))) float v2f;
typedef __attribute__((ext_vector_type(8))) float v8f;

__device__ __forceinline__ v8f wmma_f32_16x16x4(v2f a, v2f b, v8f c) {
    // 8 args: (neg_a, A, neg_b, B, c_mod, C, reuse_a, reuse_b)
    return __builtin_amdgcn_wmma_f32_16x16x4_f32(false, a, false, b, (short)0, c,
                                                 false, false);
}

// Stage 1: one wave handles one (16-query tile, M-split).  Emits partial
// (X, Y, Wsum) per query into ws[split][q][3].
// Grid layout: 32 blocks (x 8 waves = 256 query tiles) per split, so `split`
// is a scalar (blockIdx-only) value -> scalar loop bounds, SALU loop control.
__global__ void __launch_bounds__(256)
sknn_stage1(const float* __restrict__ rss, const float* __restrict__ Rrss,
            const float* __restrict__ Rloc, const float* __restrict__ sigma,
            float* __restrict__ part, int tiles_per_split) {
    const int lane  = threadIdx.x & 31;
    const int qtile = ((blockIdx.x & 31) << 3) + (threadIdx.x >> 5);
    const int split = blockIdx.x >> 5;                 // scalar per block

    const int qbase = qtile * 16;
    const int hi    = lane >> 4;               // 0: lanes 0-15, 1: lanes 16-31
    const int l15   = lane & 15;

    const float sg = sigma[0];
    const float s  = -1.4426950408889634f / (2.0f * sg * sg);   // <0, exp2 scale
    const float m2s = -2.0f * s;

    // ---- A operands (loop-invariant) --------------------------------------
    // 32-bit A 16x4 layout: lanes 0-15 hold M=0..15 with (VGPR0,VGPR1)=(K0,K1);
    // lanes 16-31 hold (K2,K3).
    const float* qp = rss + (qbase + l15) * DD;
    const float q0 = qp[0], q1 = qp[1], q2 = qp[2], q3 = qp[3], q4 = qp[4];
    const float qn = q0*q0 + q1*q1 + q2*q2 + q3*q3 + q4*q4;

    v2f A0 = { m2s * (hi ? q2 : q0), m2s * (hi ? q3 : q1) };
    // WMMA #2: lanes<16 carry (K4,K5) = (-2s*q4, s*|q|^2); lanes>=16 carry
    // (K6,K7) = (1, 0).
    v2f A1 = { hi ? 1.0f : m2s * q4, hi ? 0.0f : s * qn };

    float ax[8], ay[8], aw[8];
#pragma unroll
    for (int v = 0; v < 8; ++v) { ax[v] = 0.0f; ay[v] = 0.0f; aw[v] = 0.0f; }

    int t0 = split * tiles_per_split;                  // scalar
    int t1 = t0 + tiles_per_split; if (t1 > RT) t1 = RT;

    const float b1hi = hi ? 0.0f : 1.0f;       // K5 row of ones / K7 zeros

#pragma unroll 2
    for (int t = t0; t < t1; ++t) {
        const int r = t * 16 + l15;
        const float* bp = Rrss + r * DD;
        const float f0 = bp[0], f1 = bp[1], f2 = bp[2], f3 = bp[3], f4 = bp[4];
        const float bns = s * (f0*f0 + f1*f1 + f2*f2 + f3*f3 + f4*f4);

        // B operand 4x16: K-split mirrors A (lanes0-15: K0/K1, 16-31: K2/K3)
        v2f B0 = { hi ? f2 : f0, hi ? f3 : f1 };
        // WMMA #2: lanes<16 rows (K4,K5)=(r[4], 1); lanes>=16 rows (K6,K7)=(s*|r|^2, 0)
        v2f B1 = { hi ? bns : f4, b1hi };

        v8f c = {};                                       // inline-0 accumulator
        c = wmma_f32_16x16x4(A0, B0, c);                  // -2s * q.r (K0..3)
        c = wmma_f32_16x16x4(A1, B1, c);                  // + K4 term + norms

        const float2 loc = ((const float2*)Rloc)[r];
#pragma unroll
        for (int v = 0; v < 8; ++v) {
            float w = __builtin_amdgcn_exp2f(c[v]);       // v_exp_f32
            ax[v] = fmaf(w, loc.x, ax[v]);
            ay[v] = fmaf(w, loc.y, ay[v]);
            aw[v] += w;
        }
    }

    // Reduce across the 16 ref-columns; XOR masks 1..8 stay inside each half.
#pragma unroll
    for (int m = 1; m < 16; m <<= 1) {
#pragma unroll
        for (int v = 0; v < 8; ++v) {
            ax[v] += __shfl_xor(ax[v], m, 32);
            ay[v] += __shfl_xor(ay[v], m, 32);
            aw[v] += __shfl_xor(aw[v], m, 32);
        }
    }

    if (l15 == 0) {              // lane 0 -> queries qbase+0..7, lane 16 -> +8..15
#pragma unroll
        for (int v = 0; v < 8; ++v) {
            const int qq = qbase + v + hi * 8;
            float* p = part + ((size_t)split * NQ + qq) * 3;
            p[0] = ax[v]; p[1] = ay[v]; p[2] = aw[v];
        }
    }
}

// Stage 2: sum partials over splits, divide, emit [N,2].
__global__ void __launch_bounds__(256)
sknn_stage2(const float* __restrict__ part, float* __restrict__ out, int nsplit) {
    const int q = blockIdx.x * blockDim.x + threadIdx.x;
    if (q >= NQ) return;
    float X = 0.0f, Y = 0.0f, W = 0.0f;
    for (int s = 0; s < nsplit; ++s) {
        const float* p = part + ((size_t)s * NQ + q) * 3;
        X += p[0]; Y += p[1]; W += p[2];
    }
    const float inv = 1.0f / W;
    out[q * 2 + 0] = X * inv;
    out[q * 2 + 1] = Y * inv;
}

extern "C" void kernel_launch(void* const* d_in, const int* in_sizes, int n_in,
                              void* d_out, int out_size, void* d_ws, size_t ws_size,
                              hipStream_t stream) {
    (void)in_sizes; (void)n_in; (void)out_size;
    const float* rss   = (const float*)d_in[0];
    const float* Rrss  = (const float*)d_in[1];
    const float* Rloc  = (const float*)d_in[2];
    const float* sigma = (const float*)d_in[3];
    float* out  = (float*)d_out;
    float* part = (float*)d_ws;

    // As many M-splits as workspace allows (cap 25 -> 6400 waves in flight).
    int nsplit = 25;
    while (nsplit > 1 && (size_t)nsplit * NQ * 3 * sizeof(float) > ws_size) --nsplit;
    const int tiles_per_split = (RT + nsplit - 1) / nsplit;

    // 32 blocks per split, 8 waves per block -> 256 query tiles per split.
    sknn_stage1<<<32 * nsplit, 256, 0, stream>>>(rss, Rrss, Rloc, sigma, part,
                                                 tiles_per_split);

    sknn_stage2<<<(NQ + 255) / 256, 256, 0, stream>>>(part, out, nsplit);
}